// GraphSAGE_25958782337776
// MI455X (gfx1250) — compile-verified
//
#include <hip/hip_runtime.h>

// GraphSAGE 2-layer forward for MI455X (gfx1250, wave32).
// GEMMs: V_WMMA_F32_16X16X4_F32 (full f32 precision, matrix pipe).
// Scatter-mean: post-transform (narrow features), one wave32 per edge,
// guaranteed HW GLOBAL_ATOMIC_ADD_F32 via inline asm (no CAS-loop risk).

typedef __attribute__((ext_vector_type(2))) float v2f;
typedef __attribute__((ext_vector_type(8))) float v8f;

#define WAVES_PER_BLOCK 8

// Guaranteed hardware f32 atomic add (no-return form -> STOREcnt only).
__device__ __forceinline__ void atomic_add_f32_hw(float* p, float v)
{
    asm volatile("global_atomic_add_f32 %0, %1, off"
                 :
                 : "v"((unsigned long long)(uintptr_t)p), "v"(v)
                 : "memory");
}

// ---------------------------------------------------------------------------
// Y[N x NOUT] = X[N x K] @ W[NOUT x K]^T   (K mult of 4, NOUT mult of 16)
// One wave32 computes a 16-row x NOUT strip with WMMA f32 16x16x4.
//   A frag (16x4): lane l -> row M=l&15, k-pair khi=(l>>4)*2, float2 load.
//   B frag (4x16): lane l -> col N=l&15, same khi, float2 load from W row N.
//   C/D (16x16):   lane l -> col l&15, rows (l>>4)*8 + r, r=0..7.
// ---------------------------------------------------------------------------
template<int K, int NOUT>
__global__ __launch_bounds__(256)
void gemm_xWT_wmma(const float* __restrict__ X, const float* __restrict__ W,
                   float* __restrict__ Y, int nTiles)
{
    constexpr int NT = NOUT / 16;
    const int wave = threadIdx.x >> 5;
    const int lane = threadIdx.x & 31;
    const int tile = blockIdx.x * WAVES_PER_BLOCK + wave;
    if (tile >= nTiles) return;            // uniform per-wave: EXEC stays all-1s

    const int m   = lane & 15;
    const int khi = (lane >> 4) << 1;      // 0 or 2

    const float* xrow = X + (size_t)(tile * 16 + m) * K + khi;

    v8f acc[NT] = {};                      // zero accumulators
    const float* wrow[NT];
#pragma unroll
    for (int nt = 0; nt < NT; ++nt)
        wrow[nt] = W + (size_t)(nt * 16 + m) * K + khi;

#pragma unroll
    for (int kk = 0; kk < K; kk += 4) {
        v2f a = *(const v2f*)(xrow + kk);
#pragma unroll
        for (int nt = 0; nt < NT; ++nt) {
            v2f b = *(const v2f*)(wrow[nt] + kk);
            // 8 args: (neg_a, A, neg_b, B, c_mod, C, reuse_a, reuse_b)
            acc[nt] = __builtin_amdgcn_wmma_f32_16x16x4_f32(
                false, a, false, b, (short)0, acc[nt], false, false);
        }
    }

    const int rbase = (lane >> 4) << 3;    // 0 or 8
#pragma unroll
    for (int nt = 0; nt < NT; ++nt) {
#pragma unroll
        for (int r = 0; r < 8; ++r) {
            const int row = tile * 16 + rbase + r;
            Y[(size_t)row * NOUT + nt * 16 + m] = acc[nt][r];
        }
    }
}

// ---------------------------------------------------------------------------
// deg[dst[e]] += 1  (HW f32 atomic)
// ---------------------------------------------------------------------------
__global__ void deg_kernel(const int* __restrict__ dst, float* __restrict__ deg,
                           int nEdges)
{
    int e = blockIdx.x * blockDim.x + threadIdx.x;
    if (e < nEdges) atomic_add_f32_hw(&deg[dst[e]], 1.0f);
}

// ---------------------------------------------------------------------------
// agg[dst[e]][:] += Y[src[e]][:]  -- one wave32 per edge.
// DIM=64: lane l handles features {2l, 2l+1} (float2 gather, 2 atomics).
// DIM=32: lane l handles feature l.
// Edge indices are wave-uniform -> readfirstlane pins them to SGPRs.
// ---------------------------------------------------------------------------
template<int DIM>
__global__ __launch_bounds__(256)
void scatter_kernel(const float* __restrict__ Y,
                    const int* __restrict__ src,
                    const int* __restrict__ dst,
                    float* __restrict__ agg, int nEdges)
{
    const int lane = threadIdx.x & 31;
    const int e = blockIdx.x * WAVES_PER_BLOCK + (threadIdx.x >> 5);
    if (e >= nEdges) return;               // wave-uniform exit

    const int s = __builtin_amdgcn_readfirstlane(src[e]);
    const int d = __builtin_amdgcn_readfirstlane(dst[e]);

    if constexpr (DIM == 64) {
        const int f = lane << 1;
        v2f val = *(const v2f*)(Y + (size_t)s * 64 + f);
        float* ap = agg + (size_t)d * 64 + f;
        atomic_add_f32_hw(ap,     val.x);
        atomic_add_f32_hw(ap + 1, val.y);
    } else {
        float val = Y[(size_t)s * 32 + lane];
        atomic_add_f32_hw(agg + (size_t)d * 32 + lane, val);
    }
}

// ---------------------------------------------------------------------------
// out = [relu]( agg/max(deg,1) + bias + y_root )
// ---------------------------------------------------------------------------
template<int DIM, bool RELU>
__global__ void combine_kernel(const float* __restrict__ agg,
                               const float* __restrict__ deg,
                               const float* __restrict__ bias,
                               const float* __restrict__ yr,
                               float* __restrict__ out, int n)
{
    int tid = blockIdx.x * blockDim.x + threadIdx.x;
    if (tid >= n * DIM) return;
    int i = tid >> ((DIM == 64) ? 6 : 5);
    int f = tid & (DIM - 1);
    float v = agg[tid] / fmaxf(deg[i], 1.0f) + bias[f] + yr[tid];
    out[tid] = RELU ? fmaxf(v, 0.0f) : v;
}

// ---------------------------------------------------------------------------
extern "C" void kernel_launch(void* const* d_in, const int* in_sizes, int n_in,
                              void* d_out, int out_size, void* d_ws, size_t ws_size,
                              hipStream_t stream)
{
    const float* x   = (const float*)d_in[0];
    const int*   ei  = (const int*)d_in[1];   // [2, E] int32
    const float* W1l = (const float*)d_in[2]; // [64,128]
    const float* b1  = (const float*)d_in[3]; // [64]
    const float* W1r = (const float*)d_in[4]; // [64,128]
    const float* W2l = (const float*)d_in[5]; // [32,64]
    const float* b2  = (const float*)d_in[6]; // [32]
    const float* W2r = (const float*)d_in[7]; // [32,64]

    const int N = in_sizes[0] / 128;          // 100000 (divisible by 16)
    const int E = in_sizes[1] / 2;            // 1600000
    const int* src = ei;
    const int* dst = ei + E;

    // workspace layout (floats); layer-2 buffers reuse layer-1 regions
    float* yl1  = (float*)d_ws;               // N*64  x @ W1_l^T
    float* yr1  = yl1 + (size_t)N * 64;       // N*64  x @ W1_r^T
    float* agg  = yr1 + (size_t)N * 64;       // N*64  (reused as N*32 agg2)
    float* h    = agg + (size_t)N * 64;       // N*64  hidden
    float* deg  = h   + (size_t)N * 64;       // N
    float* yl2  = yl1;                        // N*32  h @ W2_l^T
    float* yr2  = yr1;                        // N*32  h @ W2_r^T
    float* agg2 = agg;                        // N*32

    const int nTiles = (N + 15) / 16;
    dim3 blk(256);
    dim3 gemmGrid((nTiles + WAVES_PER_BLOCK - 1) / WAVES_PER_BLOCK);
    dim3 edgeGrid((E + WAVES_PER_BLOCK - 1) / WAVES_PER_BLOCK);

    hipMemsetAsync(agg, 0, (size_t)N * 64 * sizeof(float), stream);
    hipMemsetAsync(deg, 0, (size_t)N * sizeof(float), stream);

    // ---- layer 1: transform, then scatter narrow (64-d) features ----
    gemm_xWT_wmma<128, 64><<<gemmGrid, blk, 0, stream>>>(x, W1l, yl1, nTiles);
    gemm_xWT_wmma<128, 64><<<gemmGrid, blk, 0, stream>>>(x, W1r, yr1, nTiles);

    deg_kernel<<<(E + 255) / 256, blk, 0, stream>>>(dst, deg, E);
    scatter_kernel<64><<<edgeGrid, blk, 0, stream>>>(yl1, src, dst, agg, E);

    combine_kernel<64, true><<<(N * 64 + 255) / 256, blk, 0, stream>>>(
        agg, deg, b1, yr1, h, N);

    // ---- layer 2 ----
    hipMemsetAsync(agg2, 0, (size_t)N * 32 * sizeof(float), stream);

    gemm_xWT_wmma<64, 32><<<gemmGrid, blk, 0, stream>>>(h, W2l, yl2, nTiles);
    gemm_xWT_wmma<64, 32><<<gemmGrid, blk, 0, stream>>>(h, W2r, yr2, nTiles);

    scatter_kernel<32><<<edgeGrid, blk, 0, stream>>>(yl2, src, dst, agg2, E);

    combine_kernel<32, false><<<(N * 32 + 255) / 256, blk, 0, stream>>>(
        agg2, deg, b2, yr2, (float*)d_out, N);
}